// FractalHilbertTokenizer_43112881717567
// MI455X (gfx1250) — compile-verified
//
#include <hip/hip_runtime.h>
#include <hip/hip_bf16.h>
#include <math.h>

// ---------------------------------------------------------------------------
// FractalHilbertTokenizer for MI455X (gfx1250, wave32).
// Outputs (concat in d_out, float):
//   tokens [32][5461][192], mask [32][5461], level_info [5461][16]
// Scratch (d_ws): gray image, level-5 tile sums, per-node features, split bits.
// MLP split decisions run on V_WMMA_F32_16X16X4_F32 (fp32 matrix core path).
// ---------------------------------------------------------------------------

#define BATCH  32
#define CHN    3
#define IMGH   512
#define IMGW   512
#define NSPLIT 1365      // nodes per batch image, levels 0..5
#define NALL   5461      // nodes per batch image, levels 0..6
#define TOKD   192       // 3 * 8 * 8

typedef __attribute__((ext_vector_type(2))) float v2f;
typedef __attribute__((ext_vector_type(8))) float v8f;

// Decode level-major node id -> (level, row, col). Returns level offset.
__device__ __forceinline__ int node_decode(int nid, int& l, int& r, int& c) {
  int base = 0, cnt = 1, lv = 0;
  while (nid >= base + cnt) { base += cnt; cnt <<= 2; ++lv; }
  int loc = nid - base;
  l = lv;
  r = loc >> lv;
  c = loc & ((1 << lv) - 1);
  return base;
}

// Hilbert-DFS emission position of node (l,r,c).
// HILBERT child visit order = [2,0,1,3]  =>  rank(q): 2->0, 0->1, 1->2, 3->3
__device__ __forceinline__ int dfs_pos(int l, int r, int c) {
  int pos = 0;
  int T = 1365;                       // subtree size of a level-1 node
  for (int i = 1; i <= l; ++i) {
    int br = (r >> (l - i)) & 1;
    int bc = (c >> (l - i)) & 1;
    int q  = (br << 1) | bc;
    int rank = (q == 2) ? 0 : (q == 0) ? 1 : (q == 1) ? 2 : 3;
    pos += 1 + rank * T;
    T = (T - 1) >> 2;                 // 1365 -> 341 -> 85 -> 21 -> 5 -> 1
  }
  return pos;
}

// ---------------------------------------------------------------------------
// 1) Gray image (channel mean) — one pass, feeds all 6 Sobel levels from L2.
// ---------------------------------------------------------------------------
__global__ void k_gray(const float* __restrict__ img, float* __restrict__ gray) {
  int tid = blockIdx.x * blockDim.x + threadIdx.x;
  const int HW = IMGH * IMGW;
  if (tid >= BATCH * HW) return;
  int b = tid / HW, p = tid - b * HW;
  const float* base = img + (size_t)b * CHN * HW + p;
  gray[tid] = (base[0] + base[HW] + base[2 * HW]) * (1.0f / 3.0f);
}

// ---------------------------------------------------------------------------
// 2) Level-5 16x16 color-tile sums (sum, sumsq) — basis for all mean/var.
// ---------------------------------------------------------------------------
__global__ void k_tilesum(const float* __restrict__ img,
                          float* __restrict__ sum5, float* __restrict__ sumsq5) {
  __shared__ float rs[256], rq[256];
  int blk = blockIdx.x;                 // b*1024 + tile
  int b = blk >> 10, t = blk & 1023;
  int tr = t >> 5, tc = t & 31;
  int tid = threadIdx.x;
  float s = 0.f, q = 0.f;
  for (int i = tid; i < 3 * 256; i += 256) {      // 3 channels * 16*16
    int ch = i >> 8, rem = i & 255;
    int y = rem >> 4, x = rem & 15;
    float v = img[(((size_t)b * CHN + ch) * IMGH + (tr * 16 + y)) * IMGW + (tc * 16 + x)];
    s += v; q += v * v;
  }
  rs[tid] = s; rq[tid] = q;
  __syncthreads();
  for (int w = 128; w > 0; w >>= 1) {
    if (tid < w) { rs[tid] += rs[tid + w]; rq[tid] += rq[tid + w]; }
    __syncthreads();
  }
  if (tid == 0) { sum5[blk] = rs[0]; sumsq5[blk] = rq[0]; }
}

// ---------------------------------------------------------------------------
// 3) Hierarchical mean/var for all 1365 nodes/levels 0..5 (one wave per node).
//    feat layout per node: [var, mean, edge, pad]
// ---------------------------------------------------------------------------
__global__ void k_meanvar(const float* __restrict__ sum5, const float* __restrict__ sumsq5,
                          float* __restrict__ feat) {
  int gid = blockIdx.x;                  // b*1365 + nid
  int b = gid / NSPLIT, nid = gid - b * NSPLIT;
  int l, r, c; node_decode(nid, l, r, c);
  int lg = 5 - l;                        // node covers (32>>l)^2 level-5 tiles
  int s = 1 << lg, m = s * s;
  int lane = threadIdx.x;
  float S = 0.f, Q = 0.f;
  for (int i = lane; i < m; i += 32) {
    int tr = (r << lg) + (i >> lg);
    int tc = (c << lg) + (i & (s - 1));
    int t = (b << 10) + (tr << 5) + tc;
    S += sum5[t]; Q += sumsq5[t];
  }
  for (int o = 16; o > 0; o >>= 1) { S += __shfl_down(S, o, 32); Q += __shfl_down(Q, o, 32); }
  if (lane == 0) {
    int h = IMGH >> l;
    float cnt = (float)(CHN * h * h);
    float mean = S / cnt;
    float var  = (Q - S * S / cnt) / (cnt - 1.0f);
    feat[(size_t)gid * 4 + 0] = var;
    feat[(size_t)gid * 4 + 1] = mean;
  }
}

// ---------------------------------------------------------------------------
// 4) Per-level Sobel edge mean with patch-local SAME zero-padding.
//    One block per patch; gray image lives in the 192MB L2 across all levels.
// ---------------------------------------------------------------------------
__global__ void k_edge(const float* __restrict__ gray, float* __restrict__ feat, int l) {
  __shared__ float red[256];
  int n = 1 << l, h = IMGH >> l, lg = 9 - l;
  int nn = n * n;
  int blk = blockIdx.x;                  // b*n*n + p
  int b = blk / nn, p = blk - b * nn;
  int pr = p >> l, pc = p & (n - 1);
  int Y0 = pr * h, X0 = pc * h;
  const float* g = gray + (size_t)b * IMGH * IMGW;
  int tid = threadIdx.x;
  float acc = 0.f;
  int hw = h * h;
  for (int i = tid; i < hw; i += 256) {
    int y = i >> lg, x = i & (h - 1);
    float v[3][3];
#pragma unroll
    for (int dy = 0; dy < 3; ++dy)
#pragma unroll
      for (int dx = 0; dx < 3; ++dx) {
        int yy = y + dy - 1, xx = x + dx - 1;
        v[dy][dx] = (yy >= 0 && yy < h && xx >= 0 && xx < h)
                        ? g[(size_t)(Y0 + yy) * IMGW + (X0 + xx)] : 0.f;
      }
    float gx = (v[0][2] + 2.f * v[1][2] + v[2][2]) - (v[0][0] + 2.f * v[1][0] + v[2][0]);
    float gy = (v[2][0] + 2.f * v[2][1] + v[2][2]) - (v[0][0] + 2.f * v[0][1] + v[0][2]);
    acc += sqrtf(gx * gx + gy * gy);
  }
  red[tid] = acc;
  __syncthreads();
  for (int w = 128; w > 0; w >>= 1) { if (tid < w) red[tid] += red[tid + w]; __syncthreads(); }
  if (tid == 0) {
    int offl = ((1 << (2 * l)) - 1) / 3;
    feat[(size_t)(b * NSPLIT + offl + p) * 4 + 2] = red[0] / (float)hw;
  }
}

// ---------------------------------------------------------------------------
// 5) WMMA MLP: 6->128->64->32->1 in fp32 via V_WMMA_F32_16X16X4_F32.
//    One wave handles 16 nodes (M=16). W1 is staged zero-padded (8x128) into
//    LDS cooperatively so layer-1 B fragments are branch-free ds loads (no
//    exec-divergent padded global loads between WMMAs). Activations are
//    transposed through per-wave LDS slices (padded strides -> bank-rotation,
//    conflict free); cross-lane LDS reuse is fenced with s_wait_dscnt 0
//    (same-wave LDS is in-order on CDNA5). sigmoid(x) >= 0.5  <=>  x >= 0.
// ---------------------------------------------------------------------------
__global__ __launch_bounds__(128) void k_mlp(
    const float* __restrict__ feat,
    const float* __restrict__ W1, const float* __restrict__ B1,
    const float* __restrict__ W2, const float* __restrict__ B2,
    const float* __restrict__ W3, const float* __restrict__ B3,
    const float* __restrict__ W4, const float* __restrict__ B4,
    int* __restrict__ split) {
  __shared__ float sW1[8][136];          // W1 zero-padded K 6->8 (stride 136:
                                         //  2-row lane-half shift = 16 banks)
  __shared__ float sF [4][16][8];        // features, K padded 6->8
  __shared__ float sA1[4][16][132];      // layer-1 acts (reused for layer-3 acts)
  __shared__ float sA2[4][16][68];       // layer-2 acts
  int w = threadIdx.x >> 5, lane = threadIdx.x & 31;
  int group = blockIdx.x * 4 + w;        // 16 nodes per group (wave-uniform)

  // Cooperative stage of padded W1 (whole block, before the compute guard).
  for (int j = threadIdx.x; j < 8 * 128; j += 128) {
    int kr = j >> 7, ncol = j & 127;
    sW1[kr][ncol] = (kr < 6) ? W1[kr * 128 + ncol] : 0.f;
  }
  __syncthreads();

  if (group * 16 < BATCH * NSPLIT) {
    // Stage feature rows [l, h, w, var, mean, edge, 0, 0]
    for (int j = lane; j < 128; j += 32) {
      int row = j >> 3, k = j & 7;
      int gid = group * 16 + row;
      int b = gid / NSPLIT, nid = gid - b * NSPLIT;
      int l, r, c; node_decode(nid, l, r, c);
      float val = 0.f;
      if (k == 0)                val = (float)l;
      else if (k == 1 || k == 2) val = (float)(IMGH >> l);
      else if (k < 6)            val = feat[(size_t)gid * 4 + (k - 3)];
      sF[w][row][k] = val;
    }
    asm volatile("s_wait_dscnt 0" ::: "memory");

    const int mrow  = lane & 15;         // A row / B,C column within tile
    const int khalf = (lane >> 4) << 1;  // K pair base: 0 (lanes 0-15) / 2 (16-31)
    const int mhigh = (lane >> 4) << 3;  // C/D row half offset
    v8f zero = {0.f, 0.f, 0.f, 0.f, 0.f, 0.f, 0.f, 0.f};

    // ---- Layer 1: [16x8] @ sW1[8x128] ----
    v8f acc1[8];
#pragma unroll
    for (int nt = 0; nt < 8; ++nt) acc1[nt] = zero;
#pragma unroll
    for (int k0 = 0; k0 < 8; k0 += 4) {
      v2f a;
      a.x = sF[w][mrow][k0 + khalf];
      a.y = sF[w][mrow][k0 + khalf + 1];
#pragma unroll
      for (int nt = 0; nt < 8; ++nt) {
        int ncol = nt * 16 + mrow;
        v2f bf;
        bf.x = sW1[k0 + khalf][ncol];
        bf.y = sW1[k0 + khalf + 1][ncol];
        acc1[nt] = __builtin_amdgcn_wmma_f32_16x16x4_f32(false, a, false, bf,
                                                         (short)0, acc1[nt], false, false);
      }
    }
#pragma unroll
    for (int nt = 0; nt < 8; ++nt) {
      int ncol = nt * 16 + mrow;
      float bias = B1[ncol];
#pragma unroll
      for (int v = 0; v < 8; ++v)
        sA1[w][mhigh + v][ncol] = fmaxf(acc1[nt][v] + bias, 0.f);
    }
    asm volatile("s_wait_dscnt 0" ::: "memory");

    // ---- Layer 2: [16x128] @ W2[128x64] ----
    v8f acc2[4];
#pragma unroll
    for (int nt = 0; nt < 4; ++nt) acc2[nt] = zero;
    for (int k0 = 0; k0 < 128; k0 += 4) {
      v2f a;
      a.x = sA1[w][mrow][k0 + khalf];
      a.y = sA1[w][mrow][k0 + khalf + 1];
#pragma unroll
      for (int nt = 0; nt < 4; ++nt) {
        int ncol = nt * 16 + mrow;
        v2f bf;
        bf.x = W2[(k0 + khalf) * 64 + ncol];
        bf.y = W2[(k0 + khalf + 1) * 64 + ncol];
        acc2[nt] = __builtin_amdgcn_wmma_f32_16x16x4_f32(false, a, false, bf,
                                                         (short)0, acc2[nt], false, false);
      }
    }
#pragma unroll
    for (int nt = 0; nt < 4; ++nt) {
      int ncol = nt * 16 + mrow;
      float bias = B2[ncol];
#pragma unroll
      for (int v = 0; v < 8; ++v)
        sA2[w][mhigh + v][ncol] = fmaxf(acc2[nt][v] + bias, 0.f);
    }
    asm volatile("s_wait_dscnt 0" ::: "memory");

    // ---- Layer 3: [16x64] @ W3[64x32] ----
    v8f acc3[2];
    acc3[0] = zero; acc3[1] = zero;
    for (int k0 = 0; k0 < 64; k0 += 4) {
      v2f a;
      a.x = sA2[w][mrow][k0 + khalf];
      a.y = sA2[w][mrow][k0 + khalf + 1];
#pragma unroll
      for (int nt = 0; nt < 2; ++nt) {
        int ncol = nt * 16 + mrow;
        v2f bf;
        bf.x = W3[(k0 + khalf) * 32 + ncol];
        bf.y = W3[(k0 + khalf + 1) * 32 + ncol];
        acc3[nt] = __builtin_amdgcn_wmma_f32_16x16x4_f32(false, a, false, bf,
                                                         (short)0, acc3[nt], false, false);
      }
    }
#pragma unroll
    for (int nt = 0; nt < 2; ++nt) {
      int ncol = nt * 16 + mrow;
      float bias = B3[ncol];
#pragma unroll
      for (int v = 0; v < 8; ++v)
        sA1[w][mhigh + v][ncol] = fmaxf(acc3[nt][v] + bias, 0.f);  // reuse sA1 [16x32]
    }
    asm volatile("s_wait_dscnt 0" ::: "memory");

    // ---- Layer 4: [16x32] @ W4[32x1]; split = (logit >= 0) ----
    if (lane < 16) {
      float s4 = B4[0];
#pragma unroll
      for (int k = 0; k < 32; ++k) s4 += sA1[w][lane][k] * W4[k];
      split[group * 16 + lane] = (s4 >= 0.f) ? 1 : 0;
    }
  }
}

// ---------------------------------------------------------------------------
// 6) Mask: alive = AND of ancestor splits; leaf keeps if (!split or level 6).
//    Written at Hilbert-DFS position.
// ---------------------------------------------------------------------------
__global__ void k_mask(const int* __restrict__ split, float* __restrict__ maskOut) {
  int tid = blockIdx.x * blockDim.x + threadIdx.x;
  if (tid >= BATCH * NALL) return;
  int b = tid / NALL, nid = tid - b * NALL;
  int l, r, c; node_decode(nid, l, r, c);
  const int* sp = split + b * NSPLIT;
  int alive = 1, rr = r, cc = c;
  for (int al = l - 1; al >= 0; --al) {
    rr >>= 1; cc >>= 1;
    int offa = ((1 << (2 * al)) - 1) / 3;
    alive &= sp[offa + (rr << al) + cc];
  }
  int m = alive;
  if (l < 6) {
    int offl = ((1 << (2 * l)) - 1) / 3;
    m = alive & (sp[offl + (r << l) + c] ^ 1);
  }
  maskOut[b * NALL + dfs_pos(l, r, c)] = (float)m;
}

// ---------------------------------------------------------------------------
// 7) Static level info [5461][16]: [level, path quadrants..., 0 pad].
// ---------------------------------------------------------------------------
__global__ void k_info(float* __restrict__ infoOut) {
  int nid = blockIdx.x * blockDim.x + threadIdx.x;
  if (nid >= NALL) return;
  int l, r, c; node_decode(nid, l, r, c);
  int pos = dfs_pos(l, r, c);
  float* o = infoOut + (size_t)pos * 16;
  o[0] = (float)l;
  for (int i = 1; i < 16; ++i) {
    if (i <= l) {
      int br = (r >> (l - i)) & 1, bc = (c >> (l - i)) & 1;
      o[i] = (float)((br << 1) | bc);
    } else {
      o[i] = 0.f;
    }
  }
}

// ---------------------------------------------------------------------------
// 8) Token gather: 8x8 center crop per node, scaled by mask, DFS-ordered,
//    coalesced writes (thread id fastest over the 192-elem token).
//    global_prefetch_b8 hides the scattered 32B image-row reads.
// ---------------------------------------------------------------------------
__global__ void k_tokens(const float* __restrict__ img, const float* __restrict__ maskOut,
                         float* __restrict__ tokOut) {
  long long tid = (long long)blockIdx.x * blockDim.x + threadIdx.x;
  const long long total = (long long)BATCH * NALL * TOKD;
  if (tid >= total) return;
  int e = (int)(tid % TOKD);
  long long t2 = tid / TOKD;
  int nid = (int)(t2 % NALL);
  int b = (int)(t2 / NALL);
  int l, r, c; node_decode(nid, l, r, c);
  int pos = dfs_pos(l, r, c);
  int ch = e >> 6, rem = e & 63, y = rem >> 3, x = rem & 7;
  int h = IMGH >> l, sh = (h - 8) >> 1;
  int Y = r * h + sh + y, X = c * h + sh + x;
  const float* src = &img[(((size_t)b * CHN + ch) * IMGH + Y) * IMGW + X];
  if (y < 7) __builtin_prefetch(src + IMGW, 0, 1);   // next crop row
  float m = maskOut[b * NALL + pos];
  tokOut[((size_t)b * NALL + pos) * TOKD + e] = (*src) * m;
}

// ---------------------------------------------------------------------------
extern "C" void kernel_launch(void* const* d_in, const int* in_sizes, int n_in,
                              void* d_out, int out_size, void* d_ws, size_t ws_size,
                              hipStream_t stream) {
  const float* img = (const float*)d_in[0];
  const float* W1  = (const float*)d_in[1];
  const float* B1  = (const float*)d_in[2];
  const float* W2  = (const float*)d_in[3];
  const float* B2  = (const float*)d_in[4];
  const float* W3  = (const float*)d_in[5];
  const float* B3  = (const float*)d_in[6];
  const float* W4  = (const float*)d_in[7];
  const float* B4  = (const float*)d_in[8];

  // Scratch carve (~33.1 MB of floats)
  float* ws     = (float*)d_ws;
  float* gray   = ws;                                            // 32*512*512
  float* sum5   = gray + (size_t)BATCH * IMGH * IMGW;            // 32*1024
  float* sumsq5 = sum5 + BATCH * 1024;                           // 32*1024
  float* feat   = sumsq5 + BATCH * 1024;                         // 43680*4
  int*   split  = (int*)(feat + (size_t)BATCH * NSPLIT * 4);     // 43680

  // Output carve: tokens | mask | info
  float* tokOut  = (float*)d_out;
  float* maskOut = tokOut + (size_t)BATCH * NALL * TOKD;
  float* infoOut = maskOut + (size_t)BATCH * NALL;

  k_gray<<<(BATCH * IMGH * IMGW + 255) / 256, 256, 0, stream>>>(img, gray);
  k_tilesum<<<BATCH * 1024, 256, 0, stream>>>(img, sum5, sumsq5);
  k_meanvar<<<BATCH * NSPLIT, 32, 0, stream>>>(sum5, sumsq5, feat);
  for (int l = 0; l <= 5; ++l)
    k_edge<<<BATCH * (1 << (2 * l)), 256, 0, stream>>>(gray, feat, l);

  int groups = (BATCH * NSPLIT) / 16;                            // 2730
  k_mlp<<<(groups + 3) / 4, 128, 0, stream>>>(feat, W1, B1, W2, B2, W3, B3, W4, B4, split);

  k_mask<<<(BATCH * NALL + 255) / 256, 256, 0, stream>>>(split, maskOut);
  k_info<<<(NALL + 63) / 64, 64, 0, stream>>>(infoOut);

  long long tot = (long long)BATCH * NALL * TOKD;
  k_tokens<<<(int)((tot + 255) / 256), 256, 0, stream>>>(img, maskOut, tokOut);
}